// PostProcess_36593121362043
// MI455X (gfx1250) — compile-verified
//
#include <hip/hip_runtime.h>
#include <hip/hip_bf16.h>
#include <stdint.h>

// ---------------------------------------------------------------------------
// RetinaNet-style postprocess for MI455X (gfx1250, wave32).
//  B=8 images, N=49104 anchors, C=90 classes, K=1000 pre-NMS cap.
//  Dominant traffic: 141MB classification read (~6us @ 23.3TB/s) -> staged
//  through LDS with the CDNA5 Tensor Data Mover (tensor_load_to_lds).
// ---------------------------------------------------------------------------

constexpr int   Bc  = 8;
constexpr int   Nc  = 49104;
constexpr int   Cc  = 90;
constexpr int   Kc  = 1000;
constexpr int   NBt = Bc * Nc;          // 392832 anchors total (divisible by 64)
constexpr int   CAP = 2048;             // candidate cap per image (>= K + ties)
constexpr int   TILE_ROWS = 64;         // anchors per TDM tile

constexpr size_t AL(size_t x) { return (x + 255) & ~(size_t)255; }
constexpr size_t OFF_SCORES = 0;
constexpr size_t OFF_LABELS = AL(OFF_SCORES + (size_t)NBt * 4);
constexpr size_t OFF_BOXES  = AL(OFF_LABELS + (size_t)NBt * 4);
constexpr size_t OFF_HIST   = AL(OFF_BOXES  + (size_t)NBt * 16);
constexpr size_t OFF_CNT    = AL(OFF_HIST   + (size_t)Bc * 3 * 2048 * 4);
constexpr size_t OFF_SEL    = AL(OFF_CNT    + (size_t)Bc * 4);
constexpr size_t OFF_CAND   = AL(OFF_SEL    + (size_t)Bc * 16);
constexpr size_t OFF_TBOX   = AL(OFF_CAND   + (size_t)Bc * CAP * 8);
constexpr size_t OFF_TSC    = AL(OFF_TBOX   + (size_t)Bc * Kc * 16);
constexpr size_t OFF_TLB    = AL(OFF_TSC    + (size_t)Bc * Kc * 4);
constexpr size_t OFF_TVD    = AL(OFF_TLB    + (size_t)Bc * Kc * 4);
constexpr size_t OFF_MASK   = AL(OFF_TVD    + (size_t)Bc * Kc * 4);
constexpr size_t OFF_KEEPW  = AL(OFF_MASK   + (size_t)Bc * Kc * 16 * 8);
constexpr size_t WS_NEEDED  = AL(OFF_KEEPW  + (size_t)Bc * 16 * 8);

typedef uint32_t v4u __attribute__((ext_vector_type(4)));
typedef int      v8i __attribute__((ext_vector_type(8)));
typedef int      v4i __attribute__((ext_vector_type(4)));

__device__ __forceinline__ uint32_t floatToKey(float f) {
  uint32_t b = __float_as_uint(f);
  return (b & 0x80000000u) ? ~b : (b | 0x80000000u);
}
__device__ __forceinline__ float keyToFloat(uint32_t k) {
  uint32_t b = (k & 0x80000000u) ? (k ^ 0x80000000u) : ~k;
  return __uint_as_float(b);
}

// ---------------------------------------------------------------------------
// K1: per-anchor class max/argmax (TDM-staged 64x90 f32 tile in LDS,
//     wave32 butterfly reduction) + box decode/clip.
// ---------------------------------------------------------------------------
__global__ void __launch_bounds__(256)
k_score_decode(const float* __restrict__ cls, const float* __restrict__ reg,
               const float* __restrict__ anc, const int* __restrict__ sizes,
               float* __restrict__ scores, int* __restrict__ labels,
               float* __restrict__ boxes)
{
  __shared__ float tile[TILE_ROWS * Cc];   // 23040 B
  const int tile0 = blockIdx.x * TILE_ROWS;

#if defined(__gfx1250__) && __has_builtin(__builtin_amdgcn_tensor_load_to_lds)
  if (threadIdx.x < 32) {
    // --- Build Tensor-DMA descriptor (D#) for a [rows x 90] f32 2-D tile ---
    uint32_t ldsOff = (uint32_t)__builtin_amdgcn_readfirstlane(
        (int)(uint32_t)(uintptr_t)&tile[0]);      // low 32b of flat = LDS offset
    uint64_t ga = (uint64_t)(uintptr_t)(cls + (size_t)tile0 * Cc);
    uint32_t gaLo = (uint32_t)__builtin_amdgcn_readfirstlane((int)(uint32_t)ga);
    uint32_t gaHi = (uint32_t)__builtin_amdgcn_readfirstlane((int)(uint32_t)(ga >> 32));
    int rem = NBt - tile0; if (rem < 0) rem = 0;
    uint32_t rows = (uint32_t)__builtin_amdgcn_readfirstlane(rem);

    v4u g0;
    g0[0] = 1u;                                   // count=1, user descriptor
    g0[1] = ldsOff;                               // lds_addr
    g0[2] = gaLo;                                 // global_addr[31:0]
    g0[3] = (gaHi & 0x01FFFFFFu) | (2u << 30);    // global_addr[56:32] | type=2

    v8i g1;
    g1[0] = (int)(2u << 16);                      // data_size = 4B, mask=0
    g1[1] = (int)((uint32_t)Cc << 16);            // tensor_dim0[15:0]=90
    g1[2] = (int)((rows & 0xFFFFu) << 16);        // dim0 hi=0 | tensor_dim1 lo
    g1[3] = (int)((rows >> 16) | ((uint32_t)Cc << 16));  // dim1 hi | tile_dim0=90
    g1[4] = TILE_ROWS;                            // tile_dim1=64, tile_dim2=0
    g1[5] = Cc;                                   // tensor_dim0_stride=90
    g1[6] = 0;                                    // stride hi | dim1_stride lo
    g1[7] = 0;                                    // dim1_stride hi

    v4i z4 = {0, 0, 0, 0};                        // groups 2/3 unused (2-D)
    v8i z8 = {0, 0, 0, 0, 0, 0, 0, 0};            // extra group (clang-23 form)
    __builtin_amdgcn_tensor_load_to_lds(g0, g1, z4, z4, z8, 0);
#if __has_builtin(__builtin_amdgcn_s_wait_tensorcnt)
    __builtin_amdgcn_s_wait_tensorcnt(0);
#else
    asm volatile("s_wait_tensorcnt 0" ::: "memory");
#endif
  }
  __syncthreads();
#else
  for (int i = threadIdx.x; i < TILE_ROWS * Cc; i += blockDim.x) {
    int g = tile0 + i / Cc;
    tile[i] = (g < NBt) ? cls[(size_t)tile0 * Cc + i] : 0.f;
  }
  __syncthreads();
#endif

  const int wave = threadIdx.x >> 5, lane = threadIdx.x & 31;
  // each wave reduces 8 rows: wave32 butterfly max/argmax (first-index ties)
  for (int rr = 0; rr < 8; ++rr) {
    int row = wave * 8 + rr;
    int g   = tile0 + row;
    float s = -3.0e38f; int ci = 0;
    for (int c = lane; c < Cc; c += 32) {
      float v = tile[row * Cc + c];
      if (v > s) { s = v; ci = c; }
    }
    for (int off = 16; off > 0; off >>= 1) {
      float os = __shfl_xor(s, off);
      int   oi = __shfl_xor(ci, off);
      if (os > s || (os == s && oi < ci)) { s = os; ci = oi; }
    }
    if (lane == 0 && g < NBt) { scores[g] = s; labels[g] = ci; }
  }

  // decode + clip: threads 0..63 handle one row each
  if (threadIdx.x < TILE_ROWS) {
    int g = tile0 + threadIdx.x;
    if (g < NBt) {
      int b = g / Nc, n = g - b * Nc;
      float H = (float)sizes[b * 2 + 0], W = (float)sizes[b * 2 + 1];
      float al = anc[n * 4 + 0], at = anc[n * 4 + 1];
      float ar = anc[n * 4 + 2], ab = anc[n * 4 + 3];
      float wa = ar - al, ha = ab - at;
      float cxa = al + wa * 0.5f, cya = at + ha * 0.5f;
      float r0 = reg[(size_t)g * 4 + 0], r1 = reg[(size_t)g * 4 + 1];
      float r2 = reg[(size_t)g * 4 + 2], r3 = reg[(size_t)g * 4 + 3];
      float cx = cxa + r0 * wa, cy = cya + r1 * ha;
      float w = wa * expf(r2), h = ha * expf(r3);
      float x1 = cx - w * 0.5f, y1 = cy - h * 0.5f;
      float x2 = x1 + w,        y2 = y1 + h;
      x1 = fminf(fmaxf(x1, 0.f), W - 1.f);  x2 = fminf(fmaxf(x2, 0.f), W - 1.f);
      y1 = fminf(fmaxf(y1, 0.f), H - 1.f);  y2 = fminf(fmaxf(y2, 0.f), H - 1.f);
      boxes[(size_t)g * 4 + 0] = x1; boxes[(size_t)g * 4 + 1] = y1;
      boxes[(size_t)g * 4 + 2] = x2; boxes[(size_t)g * 4 + 3] = y2;
    }
  }
}

// ---------------------------------------------------------------------------
// Radix-select (exact K-th largest key per image): 3 passes of 11/11/10 bits.
// ---------------------------------------------------------------------------
__global__ void k_hist(const float* __restrict__ scores,
                       const float* __restrict__ thr,
                       const uint32_t* __restrict__ sel,
                       uint32_t* __restrict__ hist,
                       int shift, uint32_t pmask, uint32_t bmask, int pass)
{
  float th = thr[0];
  int stride = gridDim.x * blockDim.x;
  for (int g = blockIdx.x * blockDim.x + threadIdx.x; g < NBt; g += stride) {
    float s = scores[g];
    if (s < th) continue;                       // only valid anchors compete
    int b = g / Nc;
    uint32_t k = floatToKey(s);
    if (pmask && (k & pmask) != sel[b * 4 + 0]) continue;
    uint32_t bk = (k >> shift) & bmask;
    atomicAdd(&hist[((size_t)b * 3 + pass) * 2048 + bk], 1u);
  }
}

__global__ void k_scan(const uint32_t* __restrict__ hist,
                       uint32_t* __restrict__ sel, int shift, int nb, int pass)
{
  if (threadIdx.x != 0) return;
  int b = blockIdx.x;
  uint32_t prefix = (pass == 0) ? 0u : sel[b * 4 + 0];
  uint32_t kneed  = (pass == 0) ? (uint32_t)Kc : sel[b * 4 + 1];
  const uint32_t* h = &hist[((size_t)b * 3 + pass) * 2048];
  uint32_t cum = 0; int chosen = 0;
  for (int bk = nb - 1; bk >= 0; --bk) {
    uint32_t c = h[bk];
    cum += c;
    if (cum >= kneed) { chosen = bk; kneed -= (cum - c); break; }
  }
  sel[b * 4 + 0] = prefix | ((uint32_t)chosen << shift);
  sel[b * 4 + 1] = kneed;
}

__global__ void k_compact(const float* __restrict__ scores,
                          const float* __restrict__ thr,
                          const uint32_t* __restrict__ sel,
                          uint32_t* __restrict__ cnt,
                          unsigned long long* __restrict__ cand)
{
  float th = thr[0];
  int stride = gridDim.x * blockDim.x;
  for (int g = blockIdx.x * blockDim.x + threadIdx.x; g < NBt; g += stride) {
    float s = scores[g];
    if (s < th) continue;
    int b = g / Nc, n = g - b * Nc;
    uint32_t k = floatToKey(s);
    if (k < sel[b * 4 + 0]) continue;           // key >= exact K-th largest
    uint32_t pos = atomicAdd(&cnt[b], 1u);
    if (pos < CAP)                              // invIdx: ties -> lower anchor first
      cand[(size_t)b * CAP + pos] =
          ((unsigned long long)k << 32) | (uint32_t)(0xFFFFFFFFu - (uint32_t)n);
  }
}

// ---------------------------------------------------------------------------
// Bitonic sort (descending) of <=2048 candidates, emit top-K per image.
// ---------------------------------------------------------------------------
__global__ void __launch_bounds__(1024)
k_sort(const unsigned long long* __restrict__ cand,
       const uint32_t* __restrict__ cnt,
       const float* __restrict__ boxes, const int* __restrict__ labels,
       float* __restrict__ tbox, float* __restrict__ tsc,
       int* __restrict__ tlb, int* __restrict__ tvd)
{
  __shared__ unsigned long long sk[CAP];        // 16 KB
  int b = blockIdx.x;
  uint32_t nc = cnt[b]; if (nc > CAP) nc = CAP;
  for (int i = threadIdx.x; i < CAP; i += blockDim.x)
    sk[i] = (i < (int)nc) ? cand[(size_t)b * CAP + i] : 0ull;
  __syncthreads();
  for (int k2 = 2; k2 <= CAP; k2 <<= 1) {
    for (int j = k2 >> 1; j > 0; j >>= 1) {
      for (int i = threadIdx.x; i < CAP; i += blockDim.x) {
        int ixj = i ^ j;
        if (ixj > i) {
          bool desc = (i & k2) == 0;
          unsigned long long a = sk[i], c = sk[ixj];
          if (desc ? (a < c) : (a > c)) { sk[i] = c; sk[ixj] = a; }
        }
      }
      __syncthreads();
    }
  }
  for (int t = threadIdx.x; t < Kc; t += blockDim.x) {
    unsigned long long comp = sk[t];
    size_t o = (size_t)b * Kc + t;
    if (comp) {
      uint32_t key = (uint32_t)(comp >> 32);
      int n = (int)(0xFFFFFFFFu - (uint32_t)comp);
      size_t g = (size_t)b * Nc + n;
      tbox[o * 4 + 0] = boxes[g * 4 + 0]; tbox[o * 4 + 1] = boxes[g * 4 + 1];
      tbox[o * 4 + 2] = boxes[g * 4 + 2]; tbox[o * 4 + 3] = boxes[g * 4 + 3];
      tsc[o] = keyToFloat(key); tlb[o] = labels[g]; tvd[o] = 1;
    } else {
      tbox[o * 4 + 0] = 0.f; tbox[o * 4 + 1] = 0.f;
      tbox[o * 4 + 2] = 0.f; tbox[o * 4 + 3] = 0.f;
      tsc[o] = 0.f; tlb[o] = 0; tvd[o] = 0;
    }
  }
}

// ---------------------------------------------------------------------------
// NMS phase 1: parallel IoU suppression bitmask  mask[i][w] (1000 x 16 u64).
// Class-offset trick (batched NMS) applied when staging boxes to LDS.
// ---------------------------------------------------------------------------
__global__ void __launch_bounds__(256)
k_mask(const float* __restrict__ tbox, const int* __restrict__ tlb,
       const int* __restrict__ sizes, const float* __restrict__ nmsth,
       unsigned long long* __restrict__ mask)
{
  __shared__ float X1[Kc], Y1[Kc], X2[Kc], Y2[Kc], AR[Kc];
  int b = blockIdx.y;
  float W = (float)sizes[b * 2 + 1], H = (float)sizes[b * 2 + 0];
  float offs = fmaxf(W, H) + 1.0f;
  for (int i = threadIdx.x; i < Kc; i += blockDim.x) {
    size_t o = (size_t)b * Kc + i;
    float off = (float)tlb[o] * offs;
    float x1 = tbox[o * 4 + 0] + off, y1 = tbox[o * 4 + 1] + off;
    float x2 = tbox[o * 4 + 2] + off, y2 = tbox[o * 4 + 3] + off;
    X1[i] = x1; Y1[i] = y1; X2[i] = x2; Y2[i] = y2;
    AR[i] = (x2 - x1) * (y2 - y1);
  }
  __syncthreads();
  float th = nmsth[0];
  int i = blockIdx.x * 16 + (threadIdx.x >> 4);   // row
  int w = threadIdx.x & 15;                       // 64-bit word
  if (i < Kc) {
    unsigned long long bits = 0;
    float x1 = X1[i], y1 = Y1[i], x2 = X2[i], y2 = Y2[i], ai = AR[i];
    for (int bj = 0; bj < 64; ++bj) {
      int j = (w << 6) + bj;
      if (j >= Kc || j <= i) continue;            // reference: ar > i
      float iw = fmaxf(fminf(x2, X2[j]) - fmaxf(x1, X1[j]), 0.f);
      float ih = fmaxf(fminf(y2, Y2[j]) - fmaxf(y1, Y1[j]), 0.f);
      float inter = iw * ih;
      float iou = inter / (ai + AR[j] - inter + 1e-9f);
      if (iou > th) bits |= 1ull << bj;
    }
    mask[((size_t)b * Kc + i) * 16 + w] = bits;
  }
}

// ---------------------------------------------------------------------------
// NMS phase 2: greedy scan by ONE wave32. mask matrix staged in LDS (128 KB).
// 16 suppression words live one-per-lane; liveness broadcast via __shfl.
// Depth-4 LDS software pipeline hides ds latency.
// ---------------------------------------------------------------------------
__global__ void __launch_bounds__(256)
k_nms_scan(const unsigned long long* __restrict__ mask,
           const int* __restrict__ tvd,
           unsigned long long* __restrict__ keepw)
{
  extern __shared__ unsigned long long lm[];      // Kc*16 u64 = 128000 B
  __shared__ unsigned long long vm[16];
  int b = blockIdx.x;
  for (int i = threadIdx.x; i < Kc * 16; i += blockDim.x)
    lm[i] = mask[(size_t)b * Kc * 16 + i];
  if (threadIdx.x < 16) {
    unsigned long long v = 0;
    for (int bj = 0; bj < 64; ++bj) {
      int s = threadIdx.x * 64 + bj;
      if (s < Kc && tvd[(size_t)b * Kc + s]) v |= 1ull << bj;
    }
    vm[threadIdx.x] = v;
  }
  __syncthreads();
  if (threadIdx.x < 32) {
    int lane = threadIdx.x;
    unsigned long long remv = 0, kw = 0;
    unsigned long long myv = (lane < 16) ? vm[lane] : 0ull;
    unsigned long long buf[4];
#pragma unroll
    for (int r = 0; r < 4; ++r) buf[r] = (lane < 16) ? lm[r * 16 + lane] : 0ull;
    for (int i = 0; i < Kc; ++i) {
      unsigned long long cur = buf[i & 3];
      int nx = i + 4;
      if (nx < Kc && lane < 16) buf[i & 3] = lm[nx * 16 + lane];
      int owner = i >> 6, bit = i & 63;
      unsigned long long rw = __shfl(remv, owner);
      unsigned long long vw = __shfl(myv, owner);
      bool keep = ((vw >> bit) & 1ull) && !((rw >> bit) & 1ull);
      if (keep && lane < 16) {
        remv |= cur;
        if (lane == owner) kw |= 1ull << bit;
      }
    }
    if (lane < 16) keepw[(size_t)b * 16 + lane] = kw;
  }
}

// ---------------------------------------------------------------------------
// Finalize: out5[B,K,5] ++ labels[B,K] ++ keep[B,K] (all written as f32).
// ---------------------------------------------------------------------------
__global__ void k_final(const float* __restrict__ tbox,
                        const float* __restrict__ tsc,
                        const int* __restrict__ tlb,
                        const unsigned long long* __restrict__ keepw,
                        const int* __restrict__ sizes,
                        const int* __restrict__ sizes_ori,
                        float* __restrict__ out)
{
  int t = blockIdx.x * blockDim.x + threadIdx.x;
  if (t >= Bc * Kc) return;
  int b = t / Kc, s = t - b * Kc;
  bool kp = (keepw[(size_t)b * 16 + (s >> 6)] >> (s & 63)) & 1ull;
  float H = (float)sizes[b * 2 + 0];
  float scale = (float)sizes_ori[b * 2 + 0] / H;
  float* o5 = out + (size_t)t * 5;
  if (kp) {
    o5[0] = tbox[(size_t)t * 4 + 0] * scale;
    o5[1] = tbox[(size_t)t * 4 + 1] * scale;
    o5[2] = tbox[(size_t)t * 4 + 2] * scale;
    o5[3] = tbox[(size_t)t * 4 + 3] * scale;
    o5[4] = tsc[t];
  } else {
    o5[0] = o5[1] = o5[2] = o5[3] = o5[4] = 0.f;
  }
  out[(size_t)Bc * Kc * 5 + t]           = kp ? (float)tlb[t] : -1.0f;
  out[(size_t)Bc * Kc * 6 + t]           = kp ? 1.0f : 0.0f;
}

// ---------------------------------------------------------------------------
extern "C" void kernel_launch(void* const* d_in, const int* in_sizes, int n_in,
                              void* d_out, int out_size, void* d_ws, size_t ws_size,
                              hipStream_t stream) {
  const float* cls   = (const float*)d_in[0];
  const float* reg   = (const float*)d_in[1];
  const float* anc   = (const float*)d_in[2];
  const int*   sz    = (const int*)d_in[3];
  const int*   szo   = (const int*)d_in[4];
  const float* sthr  = (const float*)d_in[5];
  const float* nthr  = (const float*)d_in[6];
  float* out = (float*)d_out;
  if (ws_size < WS_NEEDED) return;

  char* ws = (char*)d_ws;
  float*              scores = (float*)(ws + OFF_SCORES);
  int*                labels = (int*)(ws + OFF_LABELS);
  float*              boxes  = (float*)(ws + OFF_BOXES);
  uint32_t*           hist   = (uint32_t*)(ws + OFF_HIST);
  uint32_t*           cnt    = (uint32_t*)(ws + OFF_CNT);
  uint32_t*           sel    = (uint32_t*)(ws + OFF_SEL);
  unsigned long long* cand   = (unsigned long long*)(ws + OFF_CAND);
  float*              tbox   = (float*)(ws + OFF_TBOX);
  float*              tsc    = (float*)(ws + OFF_TSC);
  int*                tlb    = (int*)(ws + OFF_TLB);
  int*                tvd    = (int*)(ws + OFF_TVD);
  unsigned long long* maskm  = (unsigned long long*)(ws + OFF_MASK);
  unsigned long long* keepw  = (unsigned long long*)(ws + OFF_KEEPW);

  // zero histograms + candidate counters (contiguous region)
  (void)hipMemsetAsync(ws + OFF_HIST, 0, OFF_SEL - OFF_HIST, stream);

  // 1) scores/labels/decoded boxes (TDM-staged tiles)
  k_score_decode<<<NBt / TILE_ROWS, 256, 0, stream>>>(
      cls, reg, anc, sz, scores, labels, boxes);

  // 2) exact top-K threshold via 3-pass radix select
  const int      shifts[3] = {21, 10, 0};
  const int      nbkts[3]  = {2048, 2048, 1024};
  const uint32_t pmasks[3] = {0u, 0xFFE00000u, 0xFFFFFC00u};
  for (int p = 0; p < 3; ++p) {
    k_hist<<<768, 256, 0, stream>>>(scores, sthr, sel, hist,
                                    shifts[p], pmasks[p],
                                    (uint32_t)(nbkts[p] - 1), p);
    k_scan<<<Bc, 32, 0, stream>>>(hist, sel, shifts[p], nbkts[p], p);
  }

  // 3) compact candidates (key >= T), sort, emit top-K
  k_compact<<<768, 256, 0, stream>>>(scores, sthr, sel, cnt, cand);
  k_sort<<<Bc, 1024, 0, stream>>>(cand, cnt, boxes, labels, tbox, tsc, tlb, tvd);

  // 4) NMS: parallel mask build, single-wave greedy scan
  dim3 mg((Kc + 15) / 16, Bc);
  k_mask<<<mg, 256, 0, stream>>>(tbox, tlb, sz, nthr, maskm);
  k_nms_scan<<<Bc, 256, Kc * 16 * sizeof(unsigned long long), stream>>>(
      maskm, tvd, keepw);

  // 5) finalize outputs
  k_final<<<(Bc * Kc + 255) / 256, 256, 0, stream>>>(
      tbox, tsc, tlb, keepw, sz, szo, out);
}